// PFNet7_25486335934493
// MI455X (gfx1250) — compile-verified
//
#include <hip/hip_runtime.h>
#include <hip/hip_bf16.h>
#include <math.h>

typedef __attribute__((ext_vector_type(2))) float v2f;
typedef __attribute__((ext_vector_type(8))) float v8f;

#define NN   16384
#define KNN  16

// ---- workspace layout (float offsets) ----
#define OFF_S     0            // s        [NN*4]
#define OFF_H     65536        // h        [NN*32]
#define OFF_IDX   589824       // idx      [NN*16] (int32)
#define OFF_F80   851968       // feat80   [NN*80]  = [x(15)|0|mean(32)|max(32)]
#define OFF_FC    2162688      // featCat  [NN*288] = [x(15)|0|x1(256)|ids(6)|0(10)]
#define OFF_WOP   6881280      // WoP      [80*256]
#define OFF_W2AP  6901760      // W2aP     [272*32]
#define OFF_W3AP  6910464      // W3aP     [288*32]
#define OFF_W2DP  6919680      // W2dP     [32*16]
#define OFF_W3DP  6920192      // W3dP     [32*16]
#define OFF_B2DP  6920704      // b2dP     [16]
#define OFF_B3DP  6920720      // b3dP     [16]

// ---- output layout (floats) ----
#define OUT_IDS   0            // [NN*6]
#define OUT_P4    98304        // [NN*4]
#define OUT_EDGE  163840       // [2 * NN*16] (src row, then tgt row), stored as float

__device__ __forceinline__ v8f wmma4(v2f a, v2f b, v8f c) {
  // D = A(16x4 f32) * B(4x16 f32) + C(16x16 f32)
  return __builtin_amdgcn_wmma_f32_16x16x4_f32(
      /*neg_a=*/false, a, /*neg_b=*/false, b,
      /*c_mod=*/(short)0, c, /*reuse_a=*/false, /*reuse_b=*/false);
}

__device__ __forceinline__ float lrelu(float x) { return x > 0.f ? x : 0.01f * x; }

// ---------------------------------------------------------------------------
// K0: pack padded weight matrices into workspace (column-15 zero row inserted,
//     narrow output layers zero-padded to 16 columns).
// ---------------------------------------------------------------------------
__global__ __launch_bounds__(256)
void PFNet7_pack(const float* __restrict__ wo,
                 const float* __restrict__ w2a,
                 const float* __restrict__ w3a,
                 const float* __restrict__ w2d, const float* __restrict__ b2d,
                 const float* __restrict__ w3d, const float* __restrict__ b3d,
                 float* __restrict__ ws)
{
  const int tid = threadIdx.x;
  float* WoP  = ws + OFF_WOP;
  float* W2aP = ws + OFF_W2AP;
  float* W3aP = ws + OFF_W3AP;
  float* W2dP = ws + OFF_W2DP;
  float* W3dP = ws + OFF_W3DP;
  float* B2dP = ws + OFF_B2DP;
  float* B3dP = ws + OFF_B3DP;

  for (int i = tid; i < 80 * 256; i += 256) {        // wo: 79 rows -> 80 (row 15 = 0)
    int r = i >> 8, c = i & 255;
    WoP[i] = (r == 15) ? 0.f : wo[((r < 15) ? r : r - 1) * 256 + c];
  }
  for (int i = tid; i < 272 * 32; i += 256) {        // w2a: 271 rows -> 272
    int r = i >> 5, c = i & 31;
    W2aP[i] = (r == 15) ? 0.f : w2a[((r < 15) ? r : r - 1) * 32 + c];
  }
  for (int i = tid; i < 288 * 32; i += 256) {        // w3a: 277 rows -> 288
    int r = i >> 5, c = i & 31;
    float v = 0.f;
    if (r < 15)                 v = w3a[r * 32 + c];
    else if (r != 15 && r <= 277) v = w3a[(r - 1) * 32 + c];
    W3aP[i] = v;
  }
  for (int i = tid; i < 32 * 16; i += 256) {         // w2d 32x6 -> 32x16, w3d 32x4 -> 32x16
    int r = i >> 4, c = i & 15;
    W2dP[i] = (c < 6) ? w2d[r * 6 + c] : 0.f;
    W3dP[i] = (c < 4) ? w3d[r * 4 + c] : 0.f;
  }
  if (tid < 16) {
    B2dP[tid] = (tid < 6) ? b2d[tid] : 0.f;
    B3dP[tid] = (tid < 4) ? b3d[tid] : 0.f;
  }
}

// ---------------------------------------------------------------------------
// K1: per-node embeddings  s = x@ws+bs (15->4),  h = x@wh+bh (15->32)
// ---------------------------------------------------------------------------
__global__ __launch_bounds__(256)
void PFNet7_embed(const float* __restrict__ x,
                  const float* __restrict__ wsM, const float* __restrict__ bsV,
                  const float* __restrict__ whM, const float* __restrict__ bhV,
                  float* __restrict__ ws)
{
  const int n = blockIdx.x * 256 + threadIdx.x;
  float xr[15];
#pragma unroll
  for (int i = 0; i < 15; ++i) xr[i] = x[n * 15 + i];

  float* sO = ws + OFF_S;
  float* hO = ws + OFF_H;
#pragma unroll
  for (int j = 0; j < 4; ++j) {
    float a = bsV[j];
#pragma unroll
    for (int i = 0; i < 15; ++i) a = fmaf(xr[i], wsM[i * 4 + j], a);
    sO[n * 4 + j] = a;
  }
#pragma unroll
  for (int j = 0; j < 32; ++j) {
    float a = bhV[j];
#pragma unroll
    for (int i = 0; i < 15; ++i) a = fmaf(xr[i], whM[i * 32 + j], a);
    hO[n * 32 + j] = a;
  }
}

// ---------------------------------------------------------------------------
// K2: brute-force kNN in 4-D. One wave32 per query node. Per-lane sorted
// top-16 in registers over 512 candidates, then 32-way merge via LDS +
// wave shuffles. Writes idx[NN][16] and edge_index into d_out.
// ---------------------------------------------------------------------------
__global__ __launch_bounds__(256)
void PFNet7_knn(float* __restrict__ ws, float* __restrict__ out)
{
  __shared__ float smD[8][544];   // 8 waves x 32 lanes x 17 (conflict-padded)
  __shared__ int   smI[8][544];
  const int lane = threadIdx.x & 31;
  const int w    = threadIdx.x >> 5;
  const int q    = blockIdx.x * 8 + w;

  const float4* s4 = (const float4*)(ws + OFF_S);
  const float4 sq = s4[q];

  float dl[16]; int il[16];
#pragma unroll
  for (int j = 0; j < 16; ++j) { dl[j] = 3.4e38f; il[j] = -1; }

  for (int c = lane; c < NN; c += 32) {
    float4 sc = s4[c];
    float dx = sq.x - sc.x, dy = sq.y - sc.y;
    float dz = sq.z - sc.z, dw = sq.w - sc.w;
    float d = dx * dx + dy * dy + dz * dz + dw * dw;
    if (d < dl[15]) {                 // guarded sorted insertion (rare after warm-up)
      dl[15] = d; il[15] = c;
#pragma unroll
      for (int j = 15; j > 0; --j) {
        if (dl[j] < dl[j - 1]) {
          float td = dl[j]; dl[j] = dl[j - 1]; dl[j - 1] = td;
          int   ti = il[j]; il[j] = il[j - 1]; il[j - 1] = ti;
        }
      }
    }
  }

#pragma unroll
  for (int j = 0; j < 16; ++j) {
    smD[w][lane * 17 + j] = dl[j];
    smI[w][lane * 17 + j] = il[j];
  }
  __syncthreads();

  int*   idxb = (int*)(ws + OFF_IDX);
  float* esrc = out + OUT_EDGE;
  float* etgt = esrc + NN * KNN;

  int p = 0;                                   // per-lane head pointer into own list
  for (int r = 0; r < 16; ++r) {
    float v = (p < 16) ? smD[w][lane * 17 + p] : 3.4e38f;
    int bl = lane;
#pragma unroll
    for (int off = 16; off; off >>= 1) {       // wave32 argmin (deterministic tie-break)
      float ov  = __shfl_xor(v, off, 32);
      int   obl = __shfl_xor(bl, off, 32);
      if (ov < v || (ov == v && obl < bl)) { v = ov; bl = obl; }
    }
    int pw  = __shfl(p, bl, 32);
    int sel = smI[w][bl * 17 + pw];
    if (lane == bl) ++p;
    if (lane == 0) {
      idxb[q * 16 + r] = sel;
      esrc[q * 16 + r] = (float)sel;
      etgt[q * 16 + r] = (float)q;
    }
  }
}

// ---------------------------------------------------------------------------
// K3: edge gather + gaussian-weighted mean/max aggregation. One wave per
// node; lane = propagate-feature channel (P=32). Builds feat80 rows.
// ---------------------------------------------------------------------------
__global__ __launch_bounds__(256)
void PFNet7_gather(const float* __restrict__ x, float* __restrict__ ws)
{
  const int lane = threadIdx.x & 31;
  const int w    = threadIdx.x >> 5;
  const int q    = blockIdx.x * 8 + w;

  const float4* s4 = (const float4*)(ws + OFF_S);
  const float*  h  = ws + OFF_H;
  const int* idxb  = (const int*)(ws + OFF_IDX);

  const float4 sq = s4[q];
  float accm = 0.f, accx = -3.4e38f;
#pragma unroll 4
  for (int k = 0; k < 16; ++k) {
    int j = idxb[q * 16 + k];
    float4 sj = s4[j];
    float dx = sq.x - sj.x, dy = sq.y - sj.y;
    float dz = sq.z - sj.z, dw = sq.w - sj.w;
    float d2 = dx * dx + dy * dy + dz * dz + dw * dw;
    float wgt = __expf(-10.f * d2);
    float msg = h[j * 32 + lane] * wgt;
    accm += msg;
    accx = fmaxf(accx, msg);
  }
  float* f80 = ws + OFF_F80;
  f80[q * 80 + 16 + lane] = accm * (1.f / 16.f);
  f80[q * 80 + 48 + lane] = accx;
  if (lane < 16) f80[q * 80 + lane] = (lane < 15) ? x[q * 15 + lane] : 0.f;
}

// ---------------------------------------------------------------------------
// K4: encoder GEMM  x1 = leaky(feat80 @ WoP + bo)  -> featCat[:,16:272].
// Block = 16 rows; A-tile staged in LDS (stride 81); 8 waves x 2 col tiles,
// 20 K-chunks of 4 => 40 v_wmma_f32_16x16x4_f32 per wave.
// Also seeds featCat cols [0..15] with x (and col15=0) and zeros [272..287].
// ---------------------------------------------------------------------------
__global__ __launch_bounds__(256)
void PFNet7_encoder(const float* __restrict__ x, const float* __restrict__ bo,
                    float* __restrict__ ws)
{
  __shared__ float At[16 * 81];
  const int tid  = threadIdx.x;
  const int row0 = blockIdx.x * 16;
  const float* f80 = ws + OFF_F80;
  float* fc = ws + OFF_FC;

  for (int i = tid; i < 16 * 80; i += 256) {
    int r = i / 80, c = i % 80;
    At[r * 81 + c] = f80[(row0 + r) * 80 + c];
  }
  {
    int r = tid >> 4, c = tid & 15;
    fc[(row0 + r) * 288 + c]       = (c < 15) ? x[(row0 + r) * 15 + c] : 0.f;
    fc[(row0 + r) * 288 + 272 + c] = 0.f;
  }
  __syncthreads();

  const int lane = tid & 31, w = tid >> 5;
  const int m = lane & 15, kh = (lane >> 4) << 1;
  const int mhi = (lane >> 4) << 3, nc = lane & 15;
  const float* WoP = ws + OFF_WOP;

#pragma unroll
  for (int t = 0; t < 2; ++t) {
    const int cb = w * 32 + t * 16;
    v8f acc = {};
    for (int kc = 0; kc < 20; ++kc) {
      int k0 = kc * 4;
      v2f a; a.x = At[m * 81 + k0 + kh]; a.y = At[m * 81 + k0 + kh + 1];
      v2f b; b.x = WoP[(k0 + kh) * 256 + cb + nc];
             b.y = WoP[(k0 + kh + 1) * 256 + cb + nc];
      acc = wmma4(a, b, acc);
    }
    float bi = bo[cb + nc];
#pragma unroll
    for (int v = 0; v < 8; ++v) {
      int mm = v + mhi;
      fc[(row0 + mm) * 288 + 16 + cb + nc] = lrelu(acc[v] + bi);
    }
  }
}

// ---------------------------------------------------------------------------
// 32->32 hidden layer on a per-wave LDS tile (16x32, stride 33). 16 WMMAs.
// ---------------------------------------------------------------------------
__device__ __forceinline__
void PFNet7_hidden(float* __restrict__ hbw, const float* __restrict__ W,
                   const float* __restrict__ b,
                   int m, int kh, int mhi, int nc)
{
  v8f a0 = {}, a1 = {};
#pragma unroll
  for (int kc = 0; kc < 8; ++kc) {
    int k0 = kc * 4;
    v2f a;  a.x  = hbw[m * 33 + k0 + kh];        a.y  = hbw[m * 33 + k0 + kh + 1];
    v2f b0; b0.x = W[(k0 + kh) * 32 + nc];       b0.y = W[(k0 + kh + 1) * 32 + nc];
    v2f b1; b1.x = W[(k0 + kh) * 32 + 16 + nc];  b1.y = W[(k0 + kh + 1) * 32 + 16 + nc];
    a0 = wmma4(a, b0, a0);
    a1 = wmma4(a, b1, a1);
  }
  float bi0 = b[nc], bi1 = b[16 + nc];
#pragma unroll
  for (int v = 0; v < 8; ++v) {
    int mm = v + mhi;
    hbw[mm * 33 + nc]      = lrelu(a0[v] + bi0);
    hbw[mm * 33 + 16 + nc] = lrelu(a1[v] + bi1);
  }
  __syncthreads();
}

// ---------------------------------------------------------------------------
// K5: nn_id head.  [N,272]@W2aP -> 32 -> 32 -> 32 -> 16(pad of 6).
// One wave per 16-row tile; 4 waves per block.
// ---------------------------------------------------------------------------
__global__ __launch_bounds__(128)
void PFNet7_mlp2(const float* __restrict__ b2a,
                 const float* __restrict__ w2b, const float* __restrict__ b2b,
                 const float* __restrict__ w2c, const float* __restrict__ b2c,
                 float* __restrict__ ws, float* __restrict__ out)
{
  __shared__ float hb[4][16 * 33];
  const int lane = threadIdx.x & 31;
  const int w    = threadIdx.x >> 5;
  const int m = lane & 15, kh = (lane >> 4) << 1;
  const int mhi = (lane >> 4) << 3, nc = lane & 15;
  const int row0 = (blockIdx.x * 4 + w) * 16;
  float* hbw = &hb[w][0];
  float* fc = ws + OFF_FC;
  const float* W1 = ws + OFF_W2AP;

  // layer 1: K = 272 from featCat
  {
    v8f a0 = {}, a1 = {};
    for (int kc = 0; kc < 68; ++kc) {
      int k0 = kc * 4;
      v2f a;  a.x  = fc[(row0 + m) * 288 + k0 + kh];
              a.y  = fc[(row0 + m) * 288 + k0 + kh + 1];
      v2f b0; b0.x = W1[(k0 + kh) * 32 + nc];      b0.y = W1[(k0 + kh + 1) * 32 + nc];
      v2f b1; b1.x = W1[(k0 + kh) * 32 + 16 + nc]; b1.y = W1[(k0 + kh + 1) * 32 + 16 + nc];
      a0 = wmma4(a, b0, a0);
      a1 = wmma4(a, b1, a1);
    }
    float bi0 = b2a[nc], bi1 = b2a[16 + nc];
#pragma unroll
    for (int v = 0; v < 8; ++v) {
      int mm = v + mhi;
      hbw[mm * 33 + nc]      = lrelu(a0[v] + bi0);
      hbw[mm * 33 + 16 + nc] = lrelu(a1[v] + bi1);
    }
    __syncthreads();
  }

  PFNet7_hidden(hbw, w2b, b2b, m, kh, mhi, nc);
  PFNet7_hidden(hbw, w2c, b2c, m, kh, mhi, nc);

  // layer 4: 32 -> 16 (cols >=6 are zero weights); write cand_ids + featCat
  {
    const float* Wd = ws + OFF_W2DP;
    const float* Bd = ws + OFF_B2DP;
    v8f a0 = {};
#pragma unroll
    for (int kc = 0; kc < 8; ++kc) {
      int k0 = kc * 4;
      v2f a; a.x = hbw[m * 33 + k0 + kh]; a.y = hbw[m * 33 + k0 + kh + 1];
      v2f b; b.x = Wd[(k0 + kh) * 16 + nc]; b.y = Wd[(k0 + kh + 1) * 16 + nc];
      a0 = wmma4(a, b, a0);
    }
    float bi = Bd[nc];
    float* ids = out + OUT_IDS;
    if (nc < 6) {
#pragma unroll
      for (int v = 0; v < 8; ++v) {
        int mm = v + mhi;
        float t = a0[v] + bi;
        ids[(row0 + mm) * 6 + nc] = t;
        fc[(row0 + mm) * 288 + 272 + nc] = t;
      }
    }
  }
}

// ---------------------------------------------------------------------------
// K6: nn_p4 head.  [N,288]@W3aP -> 32 -> 32 -> 32 -> 16(pad of 4) + residual.
// ---------------------------------------------------------------------------
__global__ __launch_bounds__(128)
void PFNet7_mlp3(const float* __restrict__ x,
                 const float* __restrict__ b3a,
                 const float* __restrict__ w3b, const float* __restrict__ b3b,
                 const float* __restrict__ w3c, const float* __restrict__ b3c,
                 float* __restrict__ ws, float* __restrict__ out)
{
  __shared__ float hb[4][16 * 33];
  const int lane = threadIdx.x & 31;
  const int w    = threadIdx.x >> 5;
  const int m = lane & 15, kh = (lane >> 4) << 1;
  const int mhi = (lane >> 4) << 3, nc = lane & 15;
  const int row0 = (blockIdx.x * 4 + w) * 16;
  float* hbw = &hb[w][0];
  const float* fc = ws + OFF_FC;
  const float* W1 = ws + OFF_W3AP;

  // layer 1: K = 288 from featCat (includes cand_ids cols)
  {
    v8f a0 = {}, a1 = {};
    for (int kc = 0; kc < 72; ++kc) {
      int k0 = kc * 4;
      v2f a;  a.x  = fc[(row0 + m) * 288 + k0 + kh];
              a.y  = fc[(row0 + m) * 288 + k0 + kh + 1];
      v2f b0; b0.x = W1[(k0 + kh) * 32 + nc];      b0.y = W1[(k0 + kh + 1) * 32 + nc];
      v2f b1; b1.x = W1[(k0 + kh) * 32 + 16 + nc]; b1.y = W1[(k0 + kh + 1) * 32 + 16 + nc];
      a0 = wmma4(a, b0, a0);
      a1 = wmma4(a, b1, a1);
    }
    float bi0 = b3a[nc], bi1 = b3a[16 + nc];
#pragma unroll
    for (int v = 0; v < 8; ++v) {
      int mm = v + mhi;
      hbw[mm * 33 + nc]      = lrelu(a0[v] + bi0);
      hbw[mm * 33 + 16 + nc] = lrelu(a1[v] + bi1);
    }
    __syncthreads();
  }

  PFNet7_hidden(hbw, w3b, b3b, m, kh, mhi, nc);
  PFNet7_hidden(hbw, w3c, b3c, m, kh, mhi, nc);

  // layer 4: 32 -> 16 (cols >= 4 zero); residual add of x[:,3:7]
  {
    const float* Wd = ws + OFF_W3DP;
    const float* Bd = ws + OFF_B3DP;
    v8f a0 = {};
#pragma unroll
    for (int kc = 0; kc < 8; ++kc) {
      int k0 = kc * 4;
      v2f a; a.x = hbw[m * 33 + k0 + kh]; a.y = hbw[m * 33 + k0 + kh + 1];
      v2f b; b.x = Wd[(k0 + kh) * 16 + nc]; b.y = Wd[(k0 + kh + 1) * 16 + nc];
      a0 = wmma4(a, b, a0);
    }
    float bi = Bd[nc];
    float* p4 = out + OUT_P4;
    if (nc < 4) {
#pragma unroll
      for (int v = 0; v < 8; ++v) {
        int mm = v + mhi;
        p4[(row0 + mm) * 4 + nc] = x[(row0 + mm) * 15 + 3 + nc] + a0[v] + bi;
      }
    }
  }
}

// ---------------------------------------------------------------------------
extern "C" void kernel_launch(void* const* d_in, const int* in_sizes, int n_in,
                              void* d_out, int out_size, void* d_ws, size_t ws_size,
                              hipStream_t stream)
{
  (void)in_sizes; (void)n_in; (void)out_size; (void)ws_size;
  const float* x   = (const float*)d_in[0];
  const float* wsM = (const float*)d_in[1];
  const float* bs  = (const float*)d_in[2];
  const float* wh  = (const float*)d_in[3];
  const float* bh  = (const float*)d_in[4];
  const float* wo  = (const float*)d_in[5];
  const float* bo  = (const float*)d_in[6];
  const float* w2a = (const float*)d_in[7];
  const float* b2a = (const float*)d_in[8];
  const float* w2b = (const float*)d_in[9];
  const float* b2b = (const float*)d_in[10];
  const float* w2c = (const float*)d_in[11];
  const float* b2c = (const float*)d_in[12];
  const float* w2d = (const float*)d_in[13];
  const float* b2d = (const float*)d_in[14];
  const float* w3a = (const float*)d_in[15];
  const float* b3a = (const float*)d_in[16];
  const float* w3b = (const float*)d_in[17];
  const float* b3b = (const float*)d_in[18];
  const float* w3c = (const float*)d_in[19];
  const float* b3c = (const float*)d_in[20];
  const float* w3d = (const float*)d_in[21];
  const float* b3d = (const float*)d_in[22];

  float* out = (float*)d_out;
  float* ws  = (float*)d_ws;

  PFNet7_pack   <<<1,    256, 0, stream>>>(wo, w2a, w3a, w2d, b2d, w3d, b3d, ws);
  PFNet7_embed  <<<64,   256, 0, stream>>>(x, wsM, bs, wh, bh, ws);
  PFNet7_knn    <<<2048, 256, 0, stream>>>(ws, out);
  PFNet7_gather <<<2048, 256, 0, stream>>>(x, ws);
  PFNet7_encoder<<<1024, 256, 0, stream>>>(x, bo, ws);
  PFNet7_mlp2   <<<256,  128, 0, stream>>>(b2a, w2b, b2b, w2c, b2c, ws, out);
  PFNet7_mlp3   <<<256,  128, 0, stream>>>(x, b3a, w3b, b3b, w3c, b3c, ws, out);
}